// FilteredNoise_65214783423124
// MI455X (gfx1250) — compile-verified
//
#include <hip/hip_runtime.h>
#include <math.h>

// ---------------------------------------------------------------------------
// FilteredNoise for MI455X (gfx1250): per-frame FFT pipeline folded into 3
// constant matrices, executed as V_WMMA_F32_16X16X4_F32 GEMMs (16 frames per
// tile column). Constants staged once per workgroup into LDS (320KB per WGP
// on CDNA5) via async global->LDS b128 loads (ASYNCcnt); each wave processes
// 2 tiles reading A-operands from LDS, freeing VMEM for H/noise/output.
// ---------------------------------------------------------------------------

typedef __attribute__((ext_vector_type(2))) float v2f;
typedef __attribute__((ext_vector_type(8))) float v8f;
typedef __attribute__((ext_vector_type(4))) int   v4i;

#define AS1 __attribute__((address_space(1)))
#define AS3 __attribute__((address_space(3)))

#define FC      65      // input freq bins
#define M_IR    129     // IR length (2*FC-1)
#define L_FFT   192     // conv FFT length (M + 64 - 1)
#define NB      49      // bins actually used (97//2+1)
#define NY      97      // output samples per frame
#define FL      64      // frame hop / noise length
#define NROWS   98      // 2*NB interleaved re/im rows
#define RP      112     // rows padded to 7*16
#define KA      68      // K for matA padded to 17*4
#define KB      64      // K for matB (16*4)
#define KC      112     // K for matC padded to 28*4
#define NBATCH  32
#define NFRAME  4000
#define OUTLEN  256033  // (4000-1)*64 + 97
#define NTILES  8000    // 32*4000/16 frames per 16-wide tile
#define WPB     4       // waves per block (wave32)
#define TPW     2       // tiles per wave -> 1000 blocks, 4000 waves in flight

#define MATA_OFF 0
#define MATB_OFF (RP*KA)
#define MATC_OFF (RP*KA + RP*KB)
#define WS_FLOATS (RP*KA + RP*KB + RP*KC)   // 27328 floats = 106.75 KB

// ---------------------------------------------------------------------------
// Kernel 0: zero the output (harness poisons d_out; overlap-add accumulates).
// ---------------------------------------------------------------------------
__global__ void fn65214_zero(float* __restrict__ out, int nelem) {
  int stride = gridDim.x * blockDim.x;
  for (int i = blockIdx.x * blockDim.x + threadIdx.x; i < nelem; i += stride)
    out[i] = 0.0f;
}

// ---------------------------------------------------------------------------
// Kernel 1: build the three constant matrices in d_ws (double-precision trig).
//   matA[112][68]: H(65) -> interleaved [IRr_j; IRi_j] bins 0..48
//                  (folds irfft(129) . roll . hann . rfft(192)).
//   matB[112][64]: x(64) -> interleaved [Nr_j; Ni_j] (rfft(192) bins 0..48).
//   matC[112][112]: interleaved X -> y(97), g=0.01/97 (irfft(97) * gain).
//   Padding rows/cols are zero so the GEMM padding is exact.
// ---------------------------------------------------------------------------
__global__ void fn65214_build(float* __restrict__ ws) {
  const double PI = 3.14159265358979323846;
  const int totalA = RP * KA;
  const int totalB = RP * KB;
  const int totalC = RP * KC;
  const int total  = totalA + totalB + totalC;
  int stride = gridDim.x * blockDim.x;
  for (int i = blockIdx.x * blockDim.x + threadIdx.x; i < total; i += stride) {
    if (i < totalA) {
      int r = i / KA, k = i % KA;
      float v = 0.0f;
      if (r < NROWS && k < FC) {
        int j = r >> 1, im = r & 1;
        double s = 0.0;
        for (int m = 0; m < M_IR; ++m) {
          double win = 0.5 * (1.0 - cos(2.0 * PI * (double)m / (double)M_IR));
          double ck  = (k == 0) ? 1.0
                     : 2.0 * cos(2.0 * PI * (double)(k * (m - 64)) / (double)M_IR);
          double th  = 2.0 * PI * (double)(j * m) / (double)L_FFT;
          double bas = im ? -sin(th) : cos(th);
          s += bas * (win / 129.0) * ck;
        }
        v = (float)s;
      }
      ws[MATA_OFF + i] = v;
    } else if (i < totalA + totalB) {
      int ii = i - totalA;
      int r = ii / KB, t = ii % KB;
      float v = 0.0f;
      if (r < NROWS) {
        int j = r >> 1, im = r & 1;
        double th = 2.0 * PI * (double)(j * t) / (double)L_FFT;
        v = (float)(im ? -sin(th) : cos(th));
      }
      ws[MATB_OFF + ii] = v;
    } else {
      int ii = i - totalA - totalB;
      int r = ii / KC, c = ii % KC;
      float v = 0.0f;
      if (r < NY && c < NROWS) {
        int j = c >> 1, im = c & 1;
        double g = 0.01 / 97.0;
        if (j == 0) {
          v = im ? 0.0f : (float)g;
        } else {
          double th = 2.0 * PI * (double)(j * r) / (double)NY;
          v = (float)(g * 2.0 * (im ? -sin(th) : cos(th)));
        }
      }
      ws[MATC_OFF + ii] = v;
    }
  }
}

// ---------------------------------------------------------------------------
// Kernel 2: main WMMA pipeline.
//   Stage all constants into LDS once per workgroup (async global->LDS b128,
//   ASYNCcnt), then each wave32 runs TPW tiles:
//     GEMM1: acc1 = matA(112x68) @ Htile(68x16)        (17 k-steps x 7 tiles)
//     GEMM2: acc2 = matB(112x64) @ (2*noise-1)(64x16)  (16 x 7)
//     lane-local complex mult (re/im interleaved rows) -> LDS pair-major
//     GEMM3: y = matC(112x112) @ X(112x16)             (28 x 7)
//     overlap-add via global_atomic_add_f32 (<=2 commutative adds/sample).
// WMMA operand placement (wave32, 16x16x4 f32):
//   A: lane l holds row l%16, k pair (k0 + 2*(l/16), +1)  -> aligned b64
//   B: lane l holds col l%16, k pair (k0 + 2*(l/16), +1)
//   C/D: lane l holds col l%16, rows j + 8*(l/16) in v8f element j
// ---------------------------------------------------------------------------
__global__ void __launch_bounds__(WPB * 32) fn65214_main(
    const float* __restrict__ H, const float* __restrict__ noise,
    const float* __restrict__ ws, float* __restrict__ out)
{
  __shared__ __attribute__((aligned(16))) float cst[WS_FLOATS]; // 106.75 KB
  __shared__ v2f Xs[WPB * RP * 8];                              // 28 KB

  // ---- stage constants global -> LDS once per workgroup -----------------
#if __has_builtin(__builtin_amdgcn_global_load_async_to_lds_b128)
  {
    AS3 float* dst = (AS3 float*)cst;
    const AS1 float* src = (const AS1 float*)ws;
    for (int i = threadIdx.x * 4; i < WS_FLOATS; i += WPB * 32 * 4) {
      __builtin_amdgcn_global_load_async_to_lds_b128(
          (AS1 v4i*)(src + i), (AS3 v4i*)(dst + i), 0, 0);
    }
#if __has_builtin(__builtin_amdgcn_s_wait_asynccnt)
    __builtin_amdgcn_s_wait_asynccnt(0);
#else
    asm volatile("s_wait_asynccnt 0" ::: "memory");
#endif
  }
#else
  for (int i = threadIdx.x; i < WS_FLOATS; i += WPB * 32) cst[i] = ws[i];
#endif
  __syncthreads();

  const float* cA = cst + MATA_OFF;
  const float* cB = cst + MATB_OFF;
  const float* cC = cst + MATC_OFF;

  const int lane = threadIdx.x & 31;
  const int wave = threadIdx.x >> 5;
  const int n    = lane & 15;           // tile column (frame) / A-operand row
  const int h    = lane >> 4;           // lane half -> K-pair select
  v2f* xsw = &Xs[wave * (RP * 8)];

  for (int tt = 0; tt < TPW; ++tt) {
    const int tile = (blockIdx.x * WPB + wave) * TPW + tt;
    const int fb   = tile * 16 + n;     // flat frame index for this column

    v8f acc1[7], acc2[7];
#pragma unroll
    for (int r = 0; r < 7; ++r) { acc1[r] = 0.0f; acc2[r] = 0.0f; }

    // ---- GEMM1: interleaved IR bins from H ------------------------------
    const float* Hf = H + (size_t)fb * FC;
    for (int s = 0; s < KA / 4; ++s) {
      const int kk = 4 * s + 2 * h;
      v2f b;
      b.x = (kk     < FC) ? Hf[kk]     : 0.0f;  // H rows are 65 floats: b32s
      b.y = (kk + 1 < FC) ? Hf[kk + 1] : 0.0f;
#pragma unroll
      for (int rt = 0; rt < 7; ++rt) {
        v2f a = *(const v2f*)(cA + (rt * 16 + n) * KA + kk);  // ds_load_b64
        acc1[rt] = __builtin_amdgcn_wmma_f32_16x16x4_f32(
            false, a, false, b, (short)0, acc1[rt], false, false);
      }
    }

    // ---- GEMM2: interleaved noise bins from 2*u-1 -----------------------
    const float* Nf = noise + (size_t)fb * FL;
    for (int s = 0; s < KB / 4; ++s) {
      const int kk = 4 * s + 2 * h;
      v2f u = *(const v2f*)(Nf + kk);           // fb*64+kk even -> 8B aligned
      v2f b;
      b.x = 2.0f * u.x - 1.0f;
      b.y = 2.0f * u.y - 1.0f;
#pragma unroll
      for (int rt = 0; rt < 7; ++rt) {
        v2f a = *(const v2f*)(cB + (rt * 16 + n) * KB + kk);
        acc2[rt] = __builtin_amdgcn_wmma_f32_16x16x4_f32(
            false, a, false, b, (short)0, acc2[rt], false, false);
      }
    }

    // ---- X = N * IR (complex, lane-local via re/im row interleave) ------
    // Pair-major LDS layout: slot (row/2)*16 + n holds {Re,Im} -> b64 DS ops.
#pragma unroll
    for (int rt = 0; rt < 7; ++rt) {
#pragma unroll
      for (int c = 0; c < 4; ++c) {
        float rr = acc1[rt][2 * c], ri = acc1[rt][2 * c + 1];
        float nr = acc2[rt][2 * c], ni = acc2[rt][2 * c + 1];
        v2f xv;
        xv.x = nr * rr - ni * ri;
        xv.y = nr * ri + ni * rr;
        int row = rt * 16 + 2 * c + 8 * h;      // even interleaved row
        xsw[(row >> 1) * 16 + n] = xv;
      }
    }
    // Wave-local RAW/WAR through LDS: DS ops are in-order per wave (DScnt).

    // ---- GEMM3: y = matC @ X -------------------------------------------
    v8f accy[7];
#pragma unroll
    for (int r = 0; r < 7; ++r) accy[r] = 0.0f;

    for (int s = 0; s < KC / 4; ++s) {
      const int kk = 4 * s + 2 * h;
      v2f b = xsw[(kk >> 1) * 16 + n];
#pragma unroll
      for (int rt = 0; rt < 7; ++rt) {
        v2f a = *(const v2f*)(cC + (rt * 16 + n) * KC + kk);
        accy[rt] = __builtin_amdgcn_wmma_f32_16x16x4_f32(
            false, a, false, b, (short)0, accy[rt], false, false);
      }
    }

    // ---- overlap-add scatter (hop 64, frame output length 97) -----------
    const int bidx = fb / NFRAME;
    const int fidx = fb % NFRAME;
    float* obase = out + (size_t)bidx * OUTLEN + (size_t)fidx * FL;
#pragma unroll
    for (int rt = 0; rt < 7; ++rt) {
#pragma unroll
      for (int j = 0; j < 8; ++j) {
        int t = rt * 16 + j + 8 * h;
        if (t < NY) atomicAdd(obase + t, accy[rt][j]);
      }
    }
  }
}

// ---------------------------------------------------------------------------
extern "C" void kernel_launch(void* const* d_in, const int* in_sizes, int n_in,
                              void* d_out, int out_size, void* d_ws, size_t ws_size,
                              hipStream_t stream) {
  (void)in_sizes; (void)n_in; (void)ws_size;
  const float* H     = (const float*)d_in[0];   // (32, 4000, 65) f32
  const float* noise = (const float*)d_in[1];   // (32, 4000, 64) f32
  float* out = (float*)d_out;                   // (32, 256033) f32
  float* ws  = (float*)d_ws;                    // 3 constant matrices (~107 KB)

  fn65214_zero<<<2048, 256, 0, stream>>>(out, out_size);
  fn65214_build<<<128, 256, 0, stream>>>(ws);
  fn65214_main<<<NTILES / (WPB * TPW), WPB * 32, 0, stream>>>(H, noise, ws, out);
}